// Decoder_9079560864161
// MI455X (gfx1250) — compile-verified
//
#include <hip/hip_runtime.h>
#include <math.h>

// Problem constants (match reference)
#define Bsz   64
#define Tsz   32
#define Ssz   400
#define Esz   300
#define Hsz   600
#define Vsz   50000
#define VEXT  50050           // V + NUM_OOV
#define NEG_INF (-1.0e12f)

typedef __attribute__((ext_vector_type(2))) float v2f;
typedef __attribute__((ext_vector_type(8))) float v8f;

__device__ __forceinline__ void atomicMaxFloat(float* addr, float val) {
  unsigned int* ua = (unsigned int*)addr;
  unsigned int cur = __float_as_uint(*addr);
  while (__uint_as_float(cur) < val) {
    unsigned int prev = atomicCAS(ua, cur, __float_as_uint(val));
    if (prev == cur) break;
    cur = prev;
  }
}

// ---------------------------------------------------------------------------
// One GEMM "phase": acc += A[16,K] x B[K,16] for this wave's 16x16 tile.
// All strides/extents are compile-time constants, so the unrolled B-fragment
// loads lower to global_load_b32 with immediate offsets off one base pointer
// per 32-wide K tile, and the A-fragment reads are ds_load with immediates.
//   Arow = A + rowBase*LDA   (16 valid rows, no row guard needed)
//   Bcol = B + colClamped    (per-lane column pointer)
// ---------------------------------------------------------------------------
template <int K, int LDA, int LDB>
__device__ __forceinline__ void gemm_phase(const float* __restrict__ Arow,
                                           const float* __restrict__ Bcol,
                                           float* __restrict__ Asub,
                                           v8f& acc, int tid, int ln, int g)
{
  constexpr int KFULL = (K / 32) * 32;
  for (int k0 = 0; k0 < KFULL; k0 += 32) {
    __syncthreads();                       // previous tile fully consumed
    for (int idx = tid; idx < 512; idx += 256) {
      const int r = idx >> 5, kk = idx & 31;
      Asub[r * 34 + kk] = Arow[(size_t)r * LDA + k0 + kk];
      if (k0 + 32 < K)                     // prefetch next A tile
        __builtin_prefetch(&Arow[(size_t)r * LDA + k0 + 32 + kk], 0, 0);
    }
    __syncthreads();
    const float* bbase = Bcol + (size_t)(k0 + 2 * g) * LDB;
    const float* abase = Asub + ln * 34 + 2 * g;
    v2f bf[8];
#pragma unroll
    for (int u = 0; u < 8; ++u) {          // batch all B loads (imm offsets)
      bf[u].x = bbase[(size_t)(4 * u) * LDB];
      bf[u].y = bbase[(size_t)(4 * u + 1) * LDB];
    }
    if (k0 + 32 < K)                       // prefetch next B rows
      __builtin_prefetch(bbase + (size_t)32 * LDB, 0, 0);
#pragma unroll
    for (int u = 0; u < 8; ++u) {
      v2f a; a.x = abase[4 * u]; a.y = abase[4 * u + 1];
      acc = __builtin_amdgcn_wmma_f32_16x16x4_f32(
          false, a, false, bf[u], (short)0, acc, false, false);
    }
  }
  if constexpr (K % 32 != 0) {             // constexpr K tail (multiple of 4)
    constexpr int KT = K % 32;
    __syncthreads();
    for (int idx = tid; idx < 16 * KT; idx += 256) {
      const int r = idx / KT, kk = idx % KT;
      Asub[r * 34 + kk] = Arow[(size_t)r * LDA + KFULL + kk];
    }
    __syncthreads();
    const float* bbase = Bcol + (size_t)(KFULL + 2 * g) * LDB;
    const float* abase = Asub + ln * 34 + 2 * g;
#pragma unroll
    for (int u = 0; u < KT / 4; ++u) {
      v2f b;
      b.x = bbase[(size_t)(4 * u) * LDB];
      b.y = bbase[(size_t)(4 * u + 1) * LDB];
      v2f a; a.x = abase[4 * u]; a.y = abase[4 * u + 1];
      acc = __builtin_amdgcn_wmma_f32_16x16x4_f32(
          false, a, false, b, (short)0, acc, false, false);
    }
  }
}

// ---------------------------------------------------------------------------
// Fully-static two-operand GEMM:
//   C[.,N] = act( A1[.,K1] @ B1[K1,N] + A2[.,K2] @ B2[K2,N] + bias1 (+bias2) )
// Grid: x = M/16 row tiles (M always a multiple of 16), y = ceil(N/128).
// Block = 256 threads = 8 waves, one 16x16 tile per wave.
// ---------------------------------------------------------------------------
template <int N, int K1, int LDA1, int LDB1, int K2, int LDA2, int LDB2,
          int LDC, int ACT, bool BIAS2>
__launch_bounds__(256)
__global__ void k_gemm2(const float* __restrict__ A1, const float* __restrict__ A2,
                        const float* __restrict__ B1, const float* __restrict__ B2,
                        const float* __restrict__ bias1,
                        const float* __restrict__ bias2,
                        float* __restrict__ C)
{
  __shared__ float Asub[16 * 34];
  const int tid  = threadIdx.x;
  const int lane = tid & 31;
  const int wave = tid >> 5;
  const int g    = lane >> 4;
  const int ln   = lane & 15;
  const int rowBase = blockIdx.x * 16;
  const int col  = blockIdx.y * 128 + wave * 16 + ln;
  const int colC = (col < N) ? col : (N - 1);

  v8f acc = {0.f, 0.f, 0.f, 0.f, 0.f, 0.f, 0.f, 0.f};
  gemm_phase<K1, LDA1, LDB1>(A1 + (size_t)rowBase * LDA1, B1 + colC,
                             Asub, acc, tid, ln, g);
  if constexpr (K2 > 0)
    gemm_phase<K2, LDA2, LDB2>(A2 + (size_t)rowBase * LDA2, B2 + colC,
                               Asub, acc, tid, ln, g);

  float bsum = bias1[colC];
  if constexpr (BIAS2) bsum += bias2[colC];
  if (col < N) {
#pragma unroll
    for (int v = 0; v < 8; ++v) {
      const int row = rowBase + v + 8 * g;      // C/D layout: M = v + 8*half
      float val = acc[v] + bsum;
      if constexpr (ACT == 1) val = tanhf(val);
      C[(size_t)row * LDC + col] = val;
    }
  }
}

// ---------------------------------------------------------------------------
// Vocab projection + pointer merge, fused epilogue (write-once output uses
// non-temporal stores so W_log/memories stay resident in the 192MB L2).
//   logit = logit_in[64,600] @ W_log[600,V] + b_log
//   ext   = [logit | zeros(OOV)]
//   out[b,t,:] = where(ext + ptr_max == 0, -INF, ext + ptr_max)
// Grid: x = 4 row tiles, y = ceil(VEXT/128) = 392 col blocks.
// ---------------------------------------------------------------------------
__launch_bounds__(256)
__global__ void k_logits(const float* __restrict__ A,      // logit_in [64,600]
                         const float* __restrict__ W,      // W_log [600,V]
                         const float* __restrict__ bias,   // b_log [V]
                         const float* __restrict__ ptrbuf, // [B,VEXT]
                         float* __restrict__ out,          // [B,T,VEXT]
                         int t)
{
  __shared__ float Asub[16 * 34];
  const int tid  = threadIdx.x;
  const int lane = tid & 31;
  const int wave = tid >> 5;
  const int g    = lane >> 4;
  const int ln   = lane & 15;
  const int rowBase = blockIdx.x * 16;                 // batch tile
  const int col  = blockIdx.y * 128 + wave * 16 + ln;
  const int colC = (col < Vsz) ? col : (Vsz - 1);

  v8f acc = {0.f, 0.f, 0.f, 0.f, 0.f, 0.f, 0.f, 0.f};
  gemm_phase<Hsz, Hsz, Vsz>(A + (size_t)rowBase * Hsz, W + colC,
                            Asub, acc, tid, ln, g);

  const bool inV = (col < Vsz);
  const float bcol = inV ? bias[colC] : 0.f;
  if (col < VEXT) {
#pragma unroll
    for (int v = 0; v < 8; ++v) {
      const int b = rowBase + v + 8 * g;               // batch index 0..63
      float base = inV ? (acc[v] + bcol) : 0.f;        // ext columns -> 0
      float m = __builtin_nontemporal_load(&ptrbuf[(size_t)b * VEXT + col]);
      float r = base + ((m == NEG_INF) ? 0.f : m);
      r = (r == 0.f) ? NEG_INF : r;
      __builtin_nontemporal_store(r, &out[((size_t)b * Tsz + t) * VEXT + col]);
    }
  }
}

// ---------------------------------------------------------------------------
// Small elementwise / attention kernels
// ---------------------------------------------------------------------------
__global__ void k_init_state(const float* __restrict__ h0,
                             const float* __restrict__ c0,
                             float* __restrict__ h, float* __restrict__ c,
                             float* __restrict__ ctx) {
  int i = blockIdx.x * blockDim.x + threadIdx.x;
  if (i < Bsz * Hsz) { h[i] = h0[i]; c[i] = c0[i]; ctx[i] = 0.f; }
}

__global__ void k_ptr_init(float* __restrict__ p) {
  int i = blockIdx.x * blockDim.x + threadIdx.x;
  if (i < Bsz * VEXT) p[i] = NEG_INF;
}

__global__ void k_ptr_scatter(float* __restrict__ p,
                              const int* __restrict__ src,
                              const float* __restrict__ energy) {
  int i = blockIdx.x * blockDim.x + threadIdx.x;
  if (i >= Bsz * Ssz) return;
  int b = i / Ssz;
  int v = src[i];
  if (v >= 0 && v < VEXT)
    atomicMaxFloat(&p[(size_t)b * VEXT + v], energy[i]);
}

__global__ void k_ptr_reset(float* __restrict__ p, const int* __restrict__ src) {
  int i = blockIdx.x * blockDim.x + threadIdx.x;
  if (i >= Bsz * Ssz) return;
  int b = i / Ssz;
  int v = src[i];
  if (v >= 0 && v < VEXT) p[(size_t)b * VEXT + v] = NEG_INF;
}

__global__ void k_lstm(const float* __restrict__ gates,
                       float* __restrict__ c, float* __restrict__ h) {
  int i = blockIdx.x * blockDim.x + threadIdx.x;
  if (i >= Bsz * Hsz) return;
  int b = i / Hsz, j = i - b * Hsz;
  const float* gb = gates + (size_t)b * 4 * Hsz;
  float gi = gb[j], gf = gb[Hsz + j], gg = gb[2 * Hsz + j], go = gb[3 * Hsz + j];
  float si = 1.f / (1.f + expf(-gi));
  float sf = 1.f / (1.f + expf(-gf));
  float so = 1.f / (1.f + expf(-go));
  float cn = sf * c[i] + si * tanhf(gg);
  float hn = so * tanhf(cn);
  c[i] = cn; h[i] = hn;
}

// energy[b,s] = mask ? dot(h[b,:], mem[b,s,:]) : -INF. One wave per (b,s).
__launch_bounds__(256)
__global__ void k_energy(const float* __restrict__ h,
                         const float* __restrict__ mem,
                         const int* __restrict__ mask,
                         float* __restrict__ energy) {
  __shared__ float hs[Hsz];
  const int tid = threadIdx.x, lane = tid & 31, wave = tid >> 5;
  const int b  = blockIdx.x / (Ssz / 8);
  const int sb = blockIdx.x % (Ssz / 8);
  for (int j = tid; j < Hsz; j += 256) hs[j] = h[(size_t)b * Hsz + j];
  __syncthreads();
  const int s = sb * 8 + wave;
  const float* mrow = mem + ((size_t)b * Ssz + s) * Hsz;
  float sum = 0.f;
  for (int j = lane; j < Hsz; j += 32) sum += hs[j] * mrow[j];
  for (int off = 16; off > 0; off >>= 1) sum += __shfl_xor(sum, off, 32);
  if (lane == 0)
    energy[b * Ssz + s] = (mask[b * Ssz + s] == 0) ? NEG_INF : sum;
}

// softmax over S then ctx[b,:] = attn @ mem[b]. One block per batch.
__launch_bounds__(256)
__global__ void k_softmax_ctx(const float* __restrict__ energy,
                              const float* __restrict__ mem,
                              float* __restrict__ ctx) {
  __shared__ float attn[Ssz];
  __shared__ float red[256];
  const int b = blockIdx.x, tid = threadIdx.x;
  float mx = NEG_INF;
  for (int s = tid; s < Ssz; s += 256) mx = fmaxf(mx, energy[b * Ssz + s]);
  red[tid] = mx; __syncthreads();
  for (int o = 128; o > 0; o >>= 1) {
    if (tid < o) red[tid] = fmaxf(red[tid], red[tid + o]);
    __syncthreads();
  }
  mx = red[0]; __syncthreads();
  float lsum = 0.f;
  for (int s = tid; s < Ssz; s += 256) {
    float p = expf(energy[b * Ssz + s] - mx);
    attn[s] = p; lsum += p;
  }
  red[tid] = lsum; __syncthreads();
  for (int o = 128; o > 0; o >>= 1) {
    if (tid < o) red[tid] += red[tid + o];
    __syncthreads();
  }
  const float inv = 1.f / red[0];
  for (int j = tid; j < Hsz; j += 256) {
    float a = 0.f;
    const float* mb = mem + (size_t)b * Ssz * Hsz + j;
    for (int s = 0; s < Ssz; ++s) a += attn[s] * mb[(size_t)s * Hsz];
    ctx[(size_t)b * Hsz + j] = a * inv;
  }
}

// ---------------------------------------------------------------------------
// Host orchestration (graph-capture safe: all launches on `stream`)
// Workspace: ~76 MB of fp32 scratch.
// ---------------------------------------------------------------------------
extern "C" void kernel_launch(void* const* d_in, const int* in_sizes, int n_in,
                              void* d_out, int out_size, void* d_ws, size_t ws_size,
                              hipStream_t stream) {
  const float* trg   = (const float*)d_in[0];
  const int*   ext   = (const int*)  d_in[1];
  const float* h0    = (const float*)d_in[2];
  const float* c0    = (const float*)d_in[3];
  const float* enc   = (const float*)d_in[4];
  const int*   mask  = (const int*)  d_in[5];
  const float* W_enc = (const float*)d_in[6];
  const float* b_enc = (const float*)d_in[7];
  const float* W_red = (const float*)d_in[8];
  const float* b_red = (const float*)d_in[9];
  const float* W_ih  = (const float*)d_in[10];
  const float* W_hh  = (const float*)d_in[11];
  const float* b_ih  = (const float*)d_in[12];
  const float* b_hh  = (const float*)d_in[13];
  const float* W_cat = (const float*)d_in[14];
  const float* b_cat = (const float*)d_in[15];
  const float* W_log = (const float*)d_in[16];
  const float* b_log = (const float*)d_in[17];
  float* out = (float*)d_out;

  float* ws = (float*)d_ws;
  float* memories = ws; ws += (size_t)Bsz * Ssz * Hsz;   // 15.36M floats
  float* h        = ws; ws += Bsz * Hsz;
  float* c        = ws; ws += Bsz * Hsz;
  float* ctx      = ws; ws += Bsz * Hsz;
  float* x        = ws; ws += Bsz * Esz;
  float* gates    = ws; ws += Bsz * 4 * Hsz;
  float* energy   = ws; ws += Bsz * Ssz;
  float* logit_in = ws; ws += Bsz * Hsz;
  float* ptrbuf   = ws; ws += (size_t)Bsz * VEXT;        // 3.20M floats

  const dim3 blk(256);
  k_init_state<<<dim3((Bsz * Hsz + 255) / 256), blk, 0, stream>>>(h0, c0, h, c, ctx);
  k_ptr_init<<<dim3((Bsz * VEXT + 255) / 256), blk, 0, stream>>>(ptrbuf);

  // memories = enc @ W_enc + b_enc  (M=B*S=25600, N=H, K=H)
  k_gemm2<Hsz, Hsz, Hsz, Hsz, 0, 1, 1, Hsz, 0, false>
      <<<dim3((Bsz * Ssz) / 16, (Hsz + 127) / 128), blk, 0, stream>>>(
      enc, enc, W_enc, W_enc, b_enc, b_enc, memories);

  for (int t = 0; t < Tsz; ++t) {
    // x = [emb_t | ctx] @ W_red + b_red  (M=64, N=300, K=300+600)
    k_gemm2<Esz, Esz, Tsz * Esz, Esz, Hsz, Hsz, Esz, Esz, 0, false>
        <<<dim3(Bsz / 16, (Esz + 127) / 128), blk, 0, stream>>>(
        trg + (size_t)t * Esz, ctx,
        W_red, W_red + (size_t)Esz * Esz, b_red, b_red, x);
    // gates = x@W_ih + h@W_hh + b_ih + b_hh  (M=64, N=2400, K=300+600)
    k_gemm2<4 * Hsz, Esz, Esz, 4 * Hsz, Hsz, Hsz, 4 * Hsz, 4 * Hsz, 0, true>
        <<<dim3(Bsz / 16, (4 * Hsz + 127) / 128), blk, 0, stream>>>(
        x, h, W_ih, W_hh, b_ih, b_hh, gates);
    k_lstm<<<dim3((Bsz * Hsz + 255) / 256), blk, 0, stream>>>(gates, c, h);
    k_energy<<<dim3(Bsz * (Ssz / 8)), blk, 0, stream>>>(h, memories, mask, energy);
    k_softmax_ctx<<<dim3(Bsz), blk, 0, stream>>>(energy, memories, ctx);
    // logit_in = tanh([h|ctx] @ W_cat + b_cat)  (M=64, N=600, K=600+600)
    k_gemm2<Hsz, Hsz, Hsz, Hsz, Hsz, Hsz, Hsz, Hsz, 1, false>
        <<<dim3(Bsz / 16, (Hsz + 127) / 128), blk, 0, stream>>>(
        h, ctx, W_cat, W_cat + (size_t)Hsz * Hsz, b_cat, b_cat, logit_in);
    k_ptr_scatter<<<dim3((Bsz * Ssz + 255) / 256), blk, 0, stream>>>(ptrbuf, ext, energy);
    k_logits<<<dim3(Bsz / 16, (VEXT + 127) / 128), blk, 0, stream>>>(
        logit_in, W_log, b_log, ptrbuf, out, t);
    // reset only the touched pointer slots (keeps ptrbuf == -INF for next step)
    k_ptr_reset<<<dim3((Bsz * Ssz + 255) / 256), blk, 0, stream>>>(ptrbuf, ext);
  }
}